// DRRAA_47390669144304
// MI455X (gfx1250) — compile-verified
//
#include <hip/hip_runtime.h>
#include <math.h>

#define N_NODES 50000
#define SSAMP   5000
#define SPAD    5008      /* 313*16 */
#define NTILE   313
#define EEDGE   400000
#define EPSF    1e-6f

#define NB1     196       /* ceil(50000/256) */
#define SBLK    20        /* ceil(5000/256)  */
#define SPADBLK 20        /* ceil(5008/256)  */
#define P1_BLOCKS 512     /* 4096 waves */
#define CSPLIT  13        /* 313 row strips * 13 column phases = 4069 waves */
#define P2_BLOCKS 1563    /* ceil(400000/256) */

typedef __attribute__((ext_vector_type(2))) float v2f;
typedef __attribute__((ext_vector_type(8))) float v8f;

// ---------------- k1: softmax(Z), sigmoid(G), ZTG; per-block partials of
// colsum[8] and ZsZTG[8][8] (72 floats/block) ----------------
__global__ void k1_softmax_ztg(const float* __restrict__ Z, const float* __restrict__ G,
                               float* __restrict__ zsT, float* __restrict__ ztg,
                               float* __restrict__ part72) {
  __shared__ float lzs[256][8];
  __shared__ float lzt[256][8];
  int tid = threadIdx.x;
  int n = blockIdx.x * 256 + tid;
  float zs[8], zt[8];
  if (n < N_NODES) {
    float z[8], m = -1e30f;
    #pragma unroll
    for (int k = 0; k < 8; ++k) { z[k] = Z[k * N_NODES + n]; m = fmaxf(m, z[k]); }
    float s = 0.f;
    #pragma unroll
    for (int k = 0; k < 8; ++k) { z[k] = __expf(z[k] - m); s += z[k]; }
    float inv = 1.f / s;
    #pragma unroll
    for (int k = 0; k < 8; ++k) zs[k] = z[k] * inv;
    #pragma unroll
    for (int k = 0; k < 8; ++k) {
      float gs = 1.f / (1.f + __expf(-G[n * 8 + k]));
      zt[k] = zs[k] * gs;
    }
    #pragma unroll
    for (int k = 0; k < 8; ++k) { zsT[n*8+k] = zs[k]; ztg[n*8+k] = zt[k]; }
  } else {
    #pragma unroll
    for (int k = 0; k < 8; ++k) { zs[k] = 0.f; zt[k] = 0.f; }
  }
  #pragma unroll
  for (int k = 0; k < 8; ++k) { lzs[tid][k] = zs[k]; lzt[tid][k] = zt[k]; }
  __syncthreads();
  if (tid < 72) {
    float acc = 0.f;
    if (tid < 8) {
      for (int i = 0; i < 256; ++i) acc += lzt[i][tid];
    } else {
      int a = (tid - 8) >> 3, k = (tid - 8) & 7;
      for (int i = 0; i < 256; ++i) acc += lzs[i][a] * lzt[i][k];
    }
    part72[blockIdx.x * 72 + tid] = acc;
  }
}

// ---------------- k2: reduce 72-value partials -> consts[0..71] ----------------
__global__ void k2_reduce72(const float* __restrict__ part72, float* __restrict__ c) {
  int t = threadIdx.x;
  if (t < 72) {
    float acc = 0.f;
    for (int b = 0; b < NB1; ++b) acc += part72[b * 72 + t];
    c[t] = acc;   // [0..7]=colsum, [8..71]=ZsZTG
  }
}

// ---------------- k3: sampled B1[a][k] = sum_s Zs[a,idx_s]*ZTG[idx_s,k] partials ----------------
__global__ void k3_b1_partial(const int* __restrict__ sidx, const float* __restrict__ zsT,
                              const float* __restrict__ ztg, float* __restrict__ b1part) {
  __shared__ float lzs[256][8];
  __shared__ float lzt[256][8];
  int tid = threadIdx.x;
  int s = blockIdx.x * 256 + tid;
  if (s < SSAMP) {
    int idx = sidx[s];
    #pragma unroll
    for (int k = 0; k < 8; ++k) { lzs[tid][k] = zsT[idx*8+k]; lzt[tid][k] = ztg[idx*8+k]; }
  } else {
    #pragma unroll
    for (int k = 0; k < 8; ++k) { lzs[tid][k] = 0.f; lzt[tid][k] = 0.f; }
  }
  __syncthreads();
  if (tid < 64) {
    int a = tid >> 3, k = tid & 7;
    float acc = 0.f;
    for (int i = 0; i < 256; ++i) acc += lzs[i][a] * lzt[i][k];
    b1part[blockIdx.x * 64 + tid] = acc;
  }
}

// ---------------- k4: finalize B1, compute P = A@(B1/colsum), W = A@(ZsZTG/colsum) ----------------
// consts layout: [0..7] colsum, [8..71] ZsZTG, [72..135] B1, [136..151] P, [152..167] W
__global__ void k4_small(const float* __restrict__ b1part, const float* __restrict__ A,
                         float* __restrict__ c) {
  int t = threadIdx.x;
  if (t < 64) {
    float acc = 0.f;
    for (int b = 0; b < SBLK; ++b) acc += b1part[b * 64 + t];
    c[72 + t] = acc;
  }
  __syncthreads();
  if (t < 16) {
    int d = t >> 3, k = t & 7;
    float invc = 1.f / c[k];
    float accP = 0.f, accW = 0.f;
    #pragma unroll
    for (int a = 0; a < 8; ++a) {
      accP += A[d*8+a] * c[72 + a*8 + k];
      accW += A[d*8+a] * c[8  + a*8 + k];
    }
    c[136 + t] = accP * invc;
    c[152 + t] = accW * invc;
  }
}

// ---------------- k5: per-sample x = P @ Zs[:,idx], u = x+eps, norms, beta ----------------
__global__ void k5_samples(const int* __restrict__ sidx, const float* __restrict__ beta,
                           const float* __restrict__ zsT, const float* __restrict__ c,
                           float2* __restrict__ X2, float2* __restrict__ U2,
                           float* __restrict__ un2, float* __restrict__ xn2,
                           float* __restrict__ bs) {
  int s = blockIdx.x * 256 + threadIdx.x;
  if (s >= SPAD) return;
  if (s < SSAMP) {
    int idx = sidx[s];
    float x0 = 0.f, x1 = 0.f;
    #pragma unroll
    for (int k = 0; k < 8; ++k) {
      float z = zsT[idx*8+k];
      x0 += c[136 + k]     * z;
      x1 += c[136 + 8 + k] * z;
    }
    float u0 = x0 + EPSF, u1 = x1 + EPSF;
    X2[s] = make_float2(x0, x1);
    U2[s] = make_float2(u0, u1);
    un2[s] = u0*u0 + u1*u1;
    xn2[s] = x0*x0 + x1*x1;
    bs[s]  = beta[idx];
  } else {               // padding rows: exp(-1e30 ...) == 0, branch-free OOB
    X2[s] = make_float2(0.f, 0.f);
    U2[s] = make_float2(EPSF, EPSF);
    un2[s] = 2.f*EPSF*EPSF;
    xn2[s] = 0.f;
    bs[s]  = -1e30f;
  }
}

// ---------------- k6: per-node V = W @ Zs[:,n] ----------------
__global__ void k6_vnodes(const float* __restrict__ zsT, const float* __restrict__ c,
                          float2* __restrict__ V2) {
  int n = blockIdx.x * 256 + threadIdx.x;
  if (n >= N_NODES) return;
  float v0 = 0.f, v1 = 0.f;
  #pragma unroll
  for (int k = 0; k < 8; ++k) {
    float z = zsT[n*8+k];
    v0 += c[152 + k]     * z;
    v1 += c[152 + 8 + k] * z;
  }
  V2[n] = make_float2(v0, v1);
}

// ---------------- k7: S x S pairwise via WMMA Gram tiles ----------------
// dist^2(p,q) = |u_p|^2 + |x_q|^2 - 2 u_p.x_q  (exact rewrite of (x_p - x_q + eps))
// Each wave owns one 16-row strip (A operand + row constants hoisted into
// registers) and sweeps a strided set of column tiles: per tile only the B
// operand (one float2/lane) + two column scalars are loaded, then one
// v_wmma_f32_16x16x4_f32 + 8x (raw v_sqrt_f32 + v_exp_f32).
__global__ void k7_pairwise(const float2* __restrict__ X2, const float2* __restrict__ U2,
                            const float* __restrict__ un2, const float* __restrict__ xn2,
                            const float* __restrict__ bs, double* __restrict__ p1part) {
  __shared__ double red[256];
  int tid  = threadIdx.x;
  int lane = tid & 31;
  int wid  = (blockIdx.x * blockDim.x + tid) >> 5;   // 0..4095
  int tr   = wid / CSPLIT;                           // row strip
  int cs   = wid - tr * CSPLIT;                      // column phase
  double acc = 0.0;
  if (tr < NTILE) {                                  // wave-uniform branch
    int  p0 = tr << 4;
    int  li = lane & 15;
    bool lo = lane < 16;      // lanes 0-15 carry K=0,1; lanes 16-31 carry K=2,3 (zero)
    float2 ua = U2[p0 + li];
    v2f a; a.x = lo ? ua.x : 0.f;  a.y = lo ? ua.y : 0.f;
    int rbase = p0 + ((lane >> 4) << 3);             // C layout: M = vgpr + 8*(lane>=16)
    float rn[8], rb[8];
    #pragma unroll
    for (int v = 0; v < 8; ++v) { rn[v] = un2[rbase + v]; rb[v] = bs[rbase + v]; }
    for (int tc = cs; tc < NTILE; tc += CSPLIT) {
      int q0 = tc << 4;
      float2 xb = X2[q0 + li];
      v2f b; b.x = lo ? xb.x : 0.f;  b.y = lo ? xb.y : 0.f;
      v8f cz = {0.f,0.f,0.f,0.f,0.f,0.f,0.f,0.f};
      v8f d = __builtin_amdgcn_wmma_f32_16x16x4_f32(
          false, a, false, b, (short)0, cz, false, false);
      int   q  = q0 + li;
      float cq = xn2[q];
      float bq = bs[q];
      float fs = 0.f;
      #pragma unroll
      for (int v = 0; v < 8; ++v) {
        float d2   = fmaxf(rn[v] + cq - 2.f * d[v], 0.f);
        float dist = __builtin_amdgcn_sqrtf(d2);     // raw v_sqrt_f32, no IEEE fixup
        fs += __expf(rb[v] + bq - dist);
      }
      acc += (double)fs;
    }
  }
  red[tid] = acc;
  __syncthreads();
  for (int off = 128; off > 0; off >>= 1) {
    if (tid < off) red[tid] += red[tid + off];
    __syncthreads();
  }
  if (tid == 0) p1part[blockIdx.x] = red[0];
}

// ---------------- k8: sparse edge pass ----------------
__global__ void k8_edges(const int* __restrict__ si, const int* __restrict__ sj,
                         const float* __restrict__ beta, const float2* __restrict__ V2,
                         double* __restrict__ p2part) {
  __shared__ double red[256];
  int tid = threadIdx.x;
  int e = blockIdx.x * 256 + tid;
  double acc = 0.0;
  if (e < EEDGE) {
    int i = si[e], j = sj[e];
    float2 vi = V2[i], vj = V2[j];
    float d0 = vi.x - vj.x + EPSF;
    float d1 = vi.y - vj.y + EPSF;
    acc = (double)(beta[i] + beta[j] - __builtin_amdgcn_sqrtf(d0*d0 + d1*d1));
  }
  red[tid] = acc; __syncthreads();
  for (int off = 128; off > 0; off >>= 1) {
    if (tid < off) red[tid] += red[tid + off];
    __syncthreads();
  }
  if (tid == 0) p2part[blockIdx.x] = red[0];
}

// ---------------- k9: combine everything -> scalar ----------------
__global__ void k9_final(const double* __restrict__ p1part, const double* __restrict__ p2part,
                         const float* __restrict__ bs, float* __restrict__ out) {
  __shared__ double red[256];
  __shared__ double sums[3];
  int tid = threadIdx.x;
  // part1 full-matrix sum (includes diagonal + zero pads)
  double a = 0.0;
  for (int i = tid; i < P1_BLOCKS; i += 256) a += p1part[i];
  red[tid] = a; __syncthreads();
  for (int off = 128; off > 0; off >>= 1) { if (tid < off) red[tid] += red[tid+off]; __syncthreads(); }
  if (tid == 0) sums[0] = red[0];
  __syncthreads();
  // part2 edge sum
  a = 0.0;
  for (int i = tid; i < P2_BLOCKS; i += 256) a += p2part[i];
  red[tid] = a; __syncthreads();
  for (int off = 128; off > 0; off >>= 1) { if (tid < off) red[tid] += red[tid+off]; __syncthreads(); }
  if (tid == 0) sums[1] = red[0];
  __syncthreads();
  // diagonal: mat[p,p] = exp(2 b_p - sqrt(2)*eps)
  a = 0.0;
  const float dd = 1.41421356237f * EPSF;
  for (int s = tid; s < SSAMP; s += 256) a += (double)__expf(2.f * bs[s] - dd);
  red[tid] = a; __syncthreads();
  for (int off = 128; off > 0; off >>= 1) { if (tid < off) red[tid] += red[tid+off]; __syncthreads(); }
  if (tid == 0) sums[2] = red[0];
  __syncthreads();
  if (tid == 0) {
    const double e = 2.718281828459045;
    double z1 = 0.5 * e * e * (sums[0] - sums[2]);
    out[0] = (float)(sums[1] - z1);
  }
}

extern "C" void kernel_launch(void* const* d_in, const int* in_sizes, int n_in,
                              void* d_out, int out_size, void* d_ws, size_t ws_size,
                              hipStream_t stream) {
  (void)in_sizes; (void)n_in; (void)out_size; (void)ws_size;
  const float* beta = (const float*)d_in[0];
  const float* A    = (const float*)d_in[1];
  const float* Z    = (const float*)d_in[2];
  const float* G    = (const float*)d_in[3];
  const int* sidx   = (const int*)d_in[4];
  const int* si     = (const int*)d_in[5];
  const int* sj     = (const int*)d_in[6];
  float* out = (float*)d_out;

  char* base = (char*)d_ws;
  size_t off = 0;
  auto alloc = [&](size_t bytes) -> void* {
    void* p = base + off;
    off = (off + bytes + 255) & ~(size_t)255;
    return p;
  };
  float*  zsT    = (float*) alloc((size_t)N_NODES * 8 * 4);
  float*  ztg    = (float*) alloc((size_t)N_NODES * 8 * 4);
  float2* V2     = (float2*)alloc((size_t)N_NODES * 8);
  float2* X2     = (float2*)alloc((size_t)SPAD * 8);
  float2* U2     = (float2*)alloc((size_t)SPAD * 8);
  float*  un2    = (float*) alloc((size_t)SPAD * 4);
  float*  xn2    = (float*) alloc((size_t)SPAD * 4);
  float*  bs     = (float*) alloc((size_t)SPAD * 4);
  float*  part72 = (float*) alloc((size_t)NB1 * 72 * 4);
  float*  b1part = (float*) alloc((size_t)SBLK * 64 * 4);
  float*  consts = (float*) alloc(168 * 4);
  double* p1part = (double*)alloc((size_t)P1_BLOCKS * 8);
  double* p2part = (double*)alloc((size_t)P2_BLOCKS * 8);

  k1_softmax_ztg<<<NB1, 256, 0, stream>>>(Z, G, zsT, ztg, part72);
  k2_reduce72 <<<1, 128, 0, stream>>>(part72, consts);
  k3_b1_partial<<<SBLK, 256, 0, stream>>>(sidx, zsT, ztg, b1part);
  k4_small    <<<1, 64, 0, stream>>>(b1part, A, consts);
  k5_samples  <<<SPADBLK, 256, 0, stream>>>(sidx, beta, zsT, consts, X2, U2, un2, xn2, bs);
  k6_vnodes   <<<NB1, 256, 0, stream>>>(zsT, consts, V2);
  k7_pairwise <<<P1_BLOCKS, 256, 0, stream>>>(X2, U2, un2, xn2, bs, p1part);
  k8_edges    <<<P2_BLOCKS, 256, 0, stream>>>(si, sj, beta, V2, p2part);
  k9_final    <<<1, 256, 0, stream>>>(p1part, p2part, bs, out);
}